// ChannelMix_8005819039751
// MI455X (gfx1250) — compile-verified
//
#include <hip/hip_runtime.h>

// ChannelMix (order<=2 multiplicative channel mixing), MI455X / gfx1250.
// out[b,k,0:8,t]  = y[c,t]              (y = x with exact zeros -> 1.0)
// out[b,k,8:36,t] = y[c1,t]*y[c2,t]     (28 pairs c1<c2, lexicographic)
//
// Pair products: exact f32 8x8 outer products via V_WMMA_F32_16X16X4_F32,
// 4 t-values per WMMA (one per K-slot / D-quadrant):
//   k=0: A[0:8,0]=y(t0),   B[0,0:8]=y(t0)   -> D[0:8 ,0:8 ] = y(t0) (x) y(t0)
//   k=1: A[8:16,1]=y(t1),  B[1,8:16]=y(t1)  -> D[8:16,8:16] = y(t1) (x) y(t1)
//   k=2: A[8:16,2]=y(t2),  B[2,0:8]=y(t2)   -> D[8:16,0:8 ] = y(t2) (x) y(t2)
//   k=3: A[0:8,3]=y(t3),   B[3,8:16]=y(t3)  -> D[0:8 ,8:16] = y(t3) (x) y(t3)
// Each D entry = one RNE multiply + adds of exact 0.0 => bit-identical to v_mul.

typedef float v2f __attribute__((ext_vector_type(2)));
typedef float v8f __attribute__((ext_vector_type(8)));

namespace {
constexpr int kC       = 8;       // channels
constexpr int kT       = 512;     // time
constexpr int kComb2   = 28;      // pair rows
constexpr int kComb    = 36;      // total output rows
constexpr int kTT      = 128;     // t-tile per workgroup
constexpr int kStr     = kTT + 8; // LDS row stride (dwords): 8-bank skew per row
constexpr int kThreads = 128;     // 4 waves (wave32)
constexpr int kBK      = 8 * 32;  // B*K planes
}

__global__ __launch_bounds__(kThreads)
void channel_mix_wmma_kernel(const float* __restrict__ x,
                             float* __restrict__ out) {
  __shared__ float ybuf[kC * kStr];            //  4.3 KB
  __shared__ float pbuf[kComb2 * kStr + 32];   // 15.3 KB (+32 per-lane trash slots)

  const int wg   = blockIdx.x;            // 0 .. kBK*4 - 1
  const int bk   = wg >> 2;               // which (b,k) plane
  const int tile = (wg & 3) * kTT;        // t-tile origin

  const float* xp = x   + (size_t)bk * kC    * kT + tile;
  float*       op = out + (size_t)bk * kComb * kT + tile;

  const int tid = threadIdx.x;

  // ---- Phase 1: global -> LDS (b128 loads), exact zeros replaced by 1.0 ----
#pragma unroll
  for (int j = 0; j < 2; ++j) {
    const int v   = tid + j * kThreads;   // 0..255
    const int row = v >> 5;               // 0..7
    const int col = (v & 31) << 2;        // 0..124 step 4
    float4 q = *(const float4*)(xp + row * kT + col);
    q.x = (q.x == 0.0f) ? 1.0f : q.x;
    q.y = (q.y == 0.0f) ? 1.0f : q.y;
    q.z = (q.z == 0.0f) ? 1.0f : q.z;
    q.w = (q.w == 0.0f) ? 1.0f : q.w;
    float* yb = &ybuf[row * kStr + col];
    yb[0] = q.x; yb[1] = q.y; yb[2] = q.z; yb[3] = q.w;
  }
  __syncthreads();

  // ---- Phase 2: order-1 rows are just y; coalesced b128 stores ----
#pragma unroll
  for (int j = 0; j < 2; ++j) {
    const int v   = tid + j * kThreads;
    const int row = v >> 5;
    const int col = (v & 31) << 2;
    const float* yb = &ybuf[row * kStr + col];
    float4 q = {yb[0], yb[1], yb[2], yb[3]};
    *(float4*)(op + row * kT + col) = q;
  }

  // ---- Phase 3: pair rows, 4 t per WMMA ----
  const int wave = tid >> 5;
  const int lane = tid & 31;
  const int ch   = lane & 7;              // channel index within quadrant
  const int half = lane >> 3;             // which 8-lane group

  // Per-half assignments (derived from the A/B VGPR striping and D quadrants):
  //   half:            0     1     2     3
  //   A slot (x/y):    x     y     y     x
  //   A t offset:      0     1     3     2
  //   B slot (x/y):    x     y     x     y
  //   B t offset:      0     1     2     3
  //   D write-back t:  0     3     2     1
  const int dtA = (half == 0) ? 0 : (half == 1) ? 1 : (half == 2) ? 3 : 2;
  const int dtB = half;
  const int dtW = (half == 0) ? 0 : (half == 1) ? 3 : (half == 2) ? 2 : 1;
  const bool aInX = (half == 0) || (half == 3);
  const bool bInX = (half == 0) || (half == 2);
  const int  yoff = ch * kStr;

  const int tbase = wave * 32;
#pragma unroll 1
  for (int i = 0; i < 8; ++i) {
    const int t0 = tbase + 4 * i;

    const float va = ybuf[yoff + t0 + dtA];
    const float vb = (half < 2) ? va : ybuf[yoff + t0 + dtB];

    v2f a, b;
    a.x = aInX ? va : 0.0f;
    a.y = aInX ? 0.0f : va;
    b.x = bInX ? vb : 0.0f;
    b.y = bInX ? 0.0f : vb;

    v8f cz = {};
    v8f d = __builtin_amdgcn_wmma_f32_16x16x4_f32(
        /*neg_a=*/false, a, /*neg_b=*/false, b,
        /*c_mod=*/(short)0, cz, /*reuse_a=*/false, /*reuse_b=*/false);

    // Every lane holds pair products (c1, ch) for c1<ch in d[c1], at time dtW.
    // Invalid (c1 >= ch) stores are routed to a per-lane trash slot (branch-free).
    const int tw = t0 + dtW;
#pragma unroll
    for (int c1 = 0; c1 < 7; ++c1) {
      const int base = 7 * c1 - (c1 * (c1 - 1)) / 2 - c1 - 1; // idx = base + ch
      const int addr = (c1 < ch) ? ((base + ch) * kStr + tw)
                                 : (kComb2 * kStr + lane);
      pbuf[addr] = d[c1];
    }
  }
  __syncthreads();

  // ---- Phase 4: LDS pair tile -> global, coalesced b128 stores ----
#pragma unroll
  for (int j = 0; j < 7; ++j) {
    const int v   = tid + j * kThreads;   // 0..895
    const int row = v >> 5;               // 0..27
    const int col = (v & 31) << 2;
    const float* pb = &pbuf[row * kStr + col];
    float4 q = {pb[0], pb[1], pb[2], pb[3]};
    *(float4*)(op + (kC + row) * kT + col) = q;
  }
}

extern "C" void kernel_launch(void* const* d_in, const int* in_sizes, int n_in,
                              void* d_out, int out_size, void* d_ws, size_t ws_size,
                              hipStream_t stream) {
  (void)in_sizes; (void)n_in; (void)out_size; (void)d_ws; (void)ws_size;
  const float* x = (const float*)d_in[0];   // (8,32,8,512) f32
  // d_in[1] (weight) is the fixed binary combination map; structure hardcoded.
  float* out = (float*)d_out;               // (8,32,36,512) f32
  dim3 grid(kBK * (kT / kTT));              // 1024 workgroups
  channel_mix_wmma_kernel<<<grid, kThreads, 0, stream>>>(x, out);
}